// CurveNet_32899449487860
// MI455X (gfx1250) — compile-verified
//
#include <hip/hip_runtime.h>
#include <float.h>

typedef __attribute__((ext_vector_type(2))) float v2f;
typedef __attribute__((ext_vector_type(8))) float v8f;

#define BSZ 16
#define NBR 20

__device__ __forceinline__ float leakyf(float x) { return x >= 0.f ? x : 0.01f * x; }

// Deterministic float atomic max (max is exact -> order independent).
__device__ __forceinline__ void atomicMaxF(float* addr, float val) {
  unsigned int* ua = (unsigned int*)addr;
  unsigned int old = __float_as_uint(*addr);
  while (__uint_as_float(old) < val) {
    unsigned int assumed = old;
    old = atomicCAS(ua, assumed, __float_as_uint(val));
    if (old == assumed) break;
  }
}

__device__ __forceinline__ v8f wmma4(v2f a, v2f b, v8f c) {
  return __builtin_amdgcn_wmma_f32_16x16x4_f32(false, a, false, b, (short)0, c,
                                               false, false);
}

// Guarded (edge-tile) accumulation of one 16x16 tile at (r0, c0).
__device__ v8f accum_guarded(const float* __restrict__ A,
                             const float* __restrict__ W, int M, int Kd, int N,
                             int r0, int c0) {
  const int lane = threadIdx.x;
  const int half = lane >> 4;
  const int lm = lane & 15;
  const int m = r0 + lm;
  const int n = c0 + lm;
  v8f acc = {0.f, 0.f, 0.f, 0.f, 0.f, 0.f, 0.f, 0.f};
  for (int k0 = 0; k0 < Kd; k0 += 4) {
    const int ka = k0 + 2 * half;
    v2f av, bv;
    av.x = (m < M && ka < Kd)       ? A[(size_t)m * Kd + ka]      : 0.f;
    av.y = (m < M && (ka + 1) < Kd) ? A[(size_t)m * Kd + ka + 1]  : 0.f;
    bv.x = (n < N && ka < Kd)       ? W[(size_t)ka * N + n]       : 0.f;
    bv.y = (n < N && (ka + 1) < Kd) ? W[(size_t)(ka + 1) * N + n] : 0.f;
    acc = wmma4(av, bv, acc);
  }
  return acc;
}

// ---------------------------------------------------------------------------
// fp32 GEMM via V_WMMA_F32_16X16X4_F32. One wave32 computes a 64x16 C tile
// (4 stacked 16x16 sub-tiles sharing the B fragment). Software-pipelined
// interior path with unconditional loads; guarded path for edge tiles.
// MODE 0: C = act(A@W + bias (+R));   MODE 1: leaky + atomic max-pool over NBR
// consecutive rows into pooled[row/NBR][n].
// ---------------------------------------------------------------------------
template <int MODE>
__global__ __launch_bounds__(32) void gemm_f32_wmma_k(
    const float* __restrict__ A, const float* __restrict__ W,
    const float* __restrict__ bias, const float* __restrict__ R,
    float* __restrict__ C, int M, int Kd, int N, int act, int addRes) {
  const int lane = threadIdx.x;
  const int half = lane >> 4;
  const int lm = lane & 15;
  const int row0 = blockIdx.x * 64;
  const int col0 = blockIdx.y * 16;
  const int n = col0 + lm;
  v8f acc[4];
  const bool interior = (row0 + 64 <= M) && (col0 + 16 <= N) && ((Kd & 3) == 0);
  if (interior) {
    const v8f z = {0.f, 0.f, 0.f, 0.f, 0.f, 0.f, 0.f, 0.f};
    acc[0] = z; acc[1] = z; acc[2] = z; acc[3] = z;
    const size_t rA = (size_t)(row0 + lm) * Kd;
    const size_t rowStride16 = (size_t)16 * Kd;
    // Prologue loads (k0 = 0).
    int ka = 2 * half;
    v2f bv, a0, a1, a2, a3;
    bv.x = W[(size_t)ka * N + n];
    bv.y = W[(size_t)(ka + 1) * N + n];
    a0 = *(const v2f*)(A + rA + ka);
    a1 = *(const v2f*)(A + rA + rowStride16 + ka);
    a2 = *(const v2f*)(A + rA + 2 * rowStride16 + ka);
    a3 = *(const v2f*)(A + rA + 3 * rowStride16 + ka);
    for (int k0 = 0; k0 < Kd; k0 += 4) {
      const v2f cb = bv, c0 = a0, c1 = a1, c2 = a2, c3 = a3;
      if (k0 + 4 < Kd) {  // prefetch next k-step before the WMMAs
        const int kn = k0 + 4 + 2 * half;
        bv.x = W[(size_t)kn * N + n];
        bv.y = W[(size_t)(kn + 1) * N + n];
        a0 = *(const v2f*)(A + rA + kn);
        a1 = *(const v2f*)(A + rA + rowStride16 + kn);
        a2 = *(const v2f*)(A + rA + 2 * rowStride16 + kn);
        a3 = *(const v2f*)(A + rA + 3 * rowStride16 + kn);
      }
      acc[0] = wmma4(c0, cb, acc[0]);
      acc[1] = wmma4(c1, cb, acc[1]);
      acc[2] = wmma4(c2, cb, acc[2]);
      acc[3] = wmma4(c3, cb, acc[3]);
    }
  } else {
    const v8f z = {0.f, 0.f, 0.f, 0.f, 0.f, 0.f, 0.f, 0.f};
    for (int t = 0; t < 4; ++t) {
      const int r0 = row0 + t * 16;
      acc[t] = (r0 < M) ? accum_guarded(A, W, M, Kd, N, r0, col0) : z;
    }
  }
  // Epilogue.
  if (n < N) {
    const float bb = bias ? bias[n] : 0.f;
    for (int t = 0; t < 4; ++t) {
      const int rbase = row0 + t * 16;
      if (rbase >= M) break;
#pragma unroll
      for (int r = 0; r < 8; ++r) {
        const int mr = rbase + r + 8 * half;
        if (mr < M) {
          if (MODE == 0) {
            float v = acc[t][r] + bb;
            if (addRes) v += R[(size_t)mr * N + n];
            if (act == 1) v = leakyf(v);
            else if (act == 2) v = fmaxf(v, 0.f);
            C[(size_t)mr * N + n] = v;
          } else {
            const float v = leakyf(acc[t][r] + bb);
            atomicMaxF(&C[(size_t)(mr / NBR) * N + n], v);
          }
        }
      }
    }
  }
}

// ---------------------------------------------------------------------------
// Farthest point sampling: one block (8 wave32) per batch. Matches lax.scan
// semantics (idx[0]=0; dist init 1e10; argmax picks lowest index on ties).
// wave32-aware: per-wave argmax via lane shuffles (no barriers), single
// cross-wave LDS combine -> only 2 workgroup barriers per iteration.
// ---------------------------------------------------------------------------
#define FPS_T 256
__global__ void fps_kernel(const float* __restrict__ pts, int n, int n_g,
                           int* __restrict__ outIdx) {
  const int b = blockIdx.x;
  const int tid = threadIdx.x;       // 0..255
  const int lane = tid & 31;
  const int wave = tid >> 5;         // 0..7
  const float* P = pts + (size_t)b * n * 3;
  int* out = outIdx + (size_t)b * n_g;
  const int nper = (n + FPS_T - 1) / FPS_T;  // <= 8 for n <= 2048
  float dmin[8];
  for (int t = 0; t < 8; ++t) dmin[t] = 1e10f;
  __shared__ float swv[8];
  __shared__ int swi[8];
  __shared__ int sfar;
  int far = 0;
  for (int g = 0; g < n_g; ++g) {
    if (tid == 0) out[g] = far;
    const float cx = P[far * 3 + 0], cy = P[far * 3 + 1], cz = P[far * 3 + 2];
    float bestv = -1.f;
    int besti = n;
    for (int t = 0; t < nper; ++t) {
      const int i = tid + t * FPS_T;
      if (i < n) {
        const float dx = P[i * 3 + 0] - cx;
        const float dy = P[i * 3 + 1] - cy;
        const float dz = P[i * 3 + 2] - cz;
        const float d = dx * dx + dy * dy + dz * dz;
        float dm = dmin[t];
        dm = d < dm ? d : dm;
        dmin[t] = dm;
        if (dm > bestv || (dm == bestv && i < besti)) { bestv = dm; besti = i; }
      }
    }
    // Intra-wave argmax reduction (lexicographic: larger val, then lower idx).
#pragma unroll
    for (int off = 16; off > 0; off >>= 1) {
      const float ov = __shfl_xor(bestv, off, 32);
      const int oi = __shfl_xor(besti, off, 32);
      if (ov > bestv || (ov == bestv && oi < besti)) { bestv = ov; besti = oi; }
    }
    if (lane == 0) { swv[wave] = bestv; swi[wave] = besti; }
    __syncthreads();
    if (wave == 0) {
      float v = (lane < 8) ? swv[lane] : -1.f;
      int ix = (lane < 8) ? swi[lane] : n;
#pragma unroll
      for (int off = 4; off > 0; off >>= 1) {
        const float ov = __shfl_xor(v, off, 32);
        const int oi = __shfl_xor(ix, off, 32);
        if (ov > v || (ov == v && oi < ix)) { v = ov; ix = oi; }
      }
      if (lane == 0) sfar = ix;
    }
    __syncthreads();
    far = sfar;
  }
}

__global__ void gather_pts_kernel(const float* __restrict__ pts,
                                  const int* __restrict__ idx, int n, int ng,
                                  float* __restrict__ out) {
  const long t = (long)blockIdx.x * blockDim.x + threadIdx.x;
  if (t >= (long)BSZ * ng) return;
  const int b = (int)(t / ng), j = (int)(t % ng);
  const int src = idx[(size_t)b * ng + j];
  const float* ps = pts + ((size_t)b * n + src) * 3;
  float* po = out + ((size_t)b * ng + j) * 3;
  po[0] = ps[0]; po[1] = ps[1]; po[2] = ps[2];
}

// Ball query: first NBR indices (ascending) with d<=r2, padded with first.
__global__ void ball_query_kernel(const float* __restrict__ pts,
                                  const float* __restrict__ ctr, int n, int ng,
                                  float r2, int* __restrict__ out) {
  const long t = (long)blockIdx.x * blockDim.x + threadIdx.x;
  if (t >= (long)BSZ * ng) return;
  const int b = (int)(t / ng), j = (int)(t % ng);
  const float* P = pts + (size_t)b * n * 3;
  const float* q = ctr + ((size_t)b * ng + j) * 3;
  const float qx = q[0], qy = q[1], qz = q[2];
  int buf[NBR];
  int cnt = 0;
  for (int i = 0; i < n; ++i) {
    const float dx = P[i * 3 + 0] - qx;
    const float dy = P[i * 3 + 1] - qy;
    const float dz = P[i * 3 + 2] - qz;
    const float d = dx * dx + dy * dy + dz * dz;
    if (d <= r2) {
      buf[cnt++] = i;
      if (cnt == NBR) break;
    }
  }
  const int first = cnt > 0 ? buf[0] : 0;
  for (int k = cnt; k < NBR; ++k) buf[k] = first;
  int* o = out + ((size_t)b * ng + j) * NBR;
  for (int k = 0; k < NBR; ++k) o[k] = buf[k];
}

// f_ce[b,j,c] = max_k f[b, nbr[b,j,k], c]
__global__ void ball_pool_max_kernel(const float* __restrict__ f,
                                     const int* __restrict__ nbr, int n, int ng,
                                     int C, float* __restrict__ out) {
  const long t = (long)blockIdx.x * blockDim.x + threadIdx.x;
  if (t >= (long)BSZ * ng * C) return;
  const int c = (int)(t % C);
  const long r = t / C;
  const int j = (int)(r % ng), b = (int)(r / ng);
  const int* nb = nbr + ((size_t)b * ng + j) * NBR;
  float m = -FLT_MAX;
  for (int k = 0; k < NBR; ++k) {
    const float v = f[((size_t)b * n + nb[k]) * C + c];
    m = fmaxf(m, v);
  }
  out[((size_t)b * ng + j) * C + c] = m;
}

// KNN: NBR smallest squared distances; strict-< insertion keeps lower index on
// ties (matches lax.top_k tie-breaking). Includes self (d=0).
__global__ void knn_kernel(const float* __restrict__ pts, int n,
                           int* __restrict__ out) {
  const long t = (long)blockIdx.x * blockDim.x + threadIdx.x;
  if (t >= (long)BSZ * n) return;
  const int b = (int)(t / n), i = (int)(t % n);
  const float* P = pts + (size_t)b * n * 3;
  const float qx = P[i * 3 + 0], qy = P[i * 3 + 1], qz = P[i * 3 + 2];
  float bd[NBR];
  int bi[NBR];
  for (int k = 0; k < NBR; ++k) { bd[k] = 3.0e38f; bi[k] = n; }
  for (int j = 0; j < n; ++j) {
    const float dx = P[j * 3 + 0] - qx;
    const float dy = P[j * 3 + 1] - qy;
    const float dz = P[j * 3 + 2] - qz;
    const float d = dx * dx + dy * dy + dz * dz;
    if (d < bd[NBR - 1]) {
      int pos = NBR - 1;
      while (pos > 0 && d < bd[pos - 1]) {
        bd[pos] = bd[pos - 1];
        bi[pos] = bi[pos - 1];
        --pos;
      }
      bd[pos] = d;
      bi[pos] = j;
    }
  }
  int* o = out + ((size_t)b * n + i) * NBR;
  for (int k = 0; k < NBR; ++k) o[k] = bi[k];
}

// h1[b,i,k,c] = leaky( fpre[b,nbr,c]-fpre[b,i,c] + pfeat(9) @ p2fW[:,c] + p2fB[c] )
__global__ void build_h1_kernel(const float* __restrict__ fpre,
                                const float* __restrict__ pts,
                                const int* __restrict__ nbr,
                                const float* __restrict__ W,
                                const float* __restrict__ bias, int n, int dh,
                                float* __restrict__ h1) {
  const long t = (long)blockIdx.x * blockDim.x + threadIdx.x;
  if (t >= (long)BSZ * n * NBR * dh) return;
  const int c = (int)(t % dh);
  long r = t / dh;
  const int k = (int)(r % NBR);
  r /= NBR;
  const int i = (int)(r % n), b = (int)(r / n);
  const int nb = nbr[((size_t)b * n + i) * NBR + k];
  const float* pc = pts + ((size_t)b * n + i) * 3;
  const float* pn = pts + ((size_t)b * n + nb) * 3;
  float pf[9];
  pf[0] = pc[0]; pf[1] = pc[1]; pf[2] = pc[2];
  pf[3] = pn[0]; pf[4] = pn[1]; pf[5] = pn[2];
  pf[6] = pn[0] - pc[0]; pf[7] = pn[1] - pc[1]; pf[8] = pn[2] - pc[2];
  float acc = bias[c];
#pragma unroll
  for (int d = 0; d < 9; ++d) acc += pf[d] * W[d * dh + c];
  acc += fpre[((size_t)b * n + nb) * dh + c] - fpre[((size_t)b * n + i) * dh + c];
  h1[t] = leakyf(acc);
}

__global__ void fill_kernel(float* __restrict__ p, float v, long cnt) {
  const long t = (long)blockIdx.x * blockDim.x + threadIdx.x;
  if (t < cnt) p[t] = v;
}

// g[b, 0:1024] = max over 64 pts ; g[b, 1024:2048] = mean over 64 pts
__global__ void head_pool_kernel(const float* __restrict__ x,
                                 float* __restrict__ g) {
  const long t = (long)blockIdx.x * blockDim.x + threadIdx.x;
  if (t >= (long)BSZ * 1024) return;
  const int b = (int)(t / 1024), c = (int)(t % 1024);
  float mx = -FLT_MAX, sm = 0.f;
  for (int i = 0; i < 64; ++i) {
    const float v = x[((size_t)b * 64 + i) * 1024 + c];
    mx = fmaxf(mx, v);
    sm += v;
  }
  g[(size_t)b * 2048 + c] = mx;
  g[(size_t)b * 2048 + 1024 + c] = sm * (1.0f / 64.0f);
}

static inline int ceildiv_l(long a, int b) { return (int)((a + b - 1) / b); }

extern "C" void kernel_launch(void* const* d_in, const int* in_sizes, int n_in,
                              void* d_out, int out_size, void* d_ws,
                              size_t ws_size, hipStream_t stream) {
  (void)in_sizes; (void)out_size;
  if (n_in < 81) return;

  static const int DIN[8]  = {32, 64, 64, 128, 128, 256, 256, 512};
  static const int DOUT[8] = {64, 64, 128, 128, 256, 256, 512, 512};
  static const int DH[8]   = {16, 16, 32, 32, 64, 64, 128, 128};
  static const int NG[8]   = {1024, 1024, 1024, 1024, 256, 256, 64, 64};
  static const double RG[8] = {0.05, 0.05, 0.05, 0.1, 0.1, 0.2, 0.2, 0.4};

  const float* p_in = (const float*)d_in[0];
  // pytree flatten order: cics (list, each dict keys sorted: mlp,p2f,pre,pst,
  // [shortcut]; each linear W then b), then head (fc,mlp1,mlp2), then stem.
  struct CicP {
    const float *mlpW, *mlpB, *p2fW, *p2fB, *preW, *preB, *pstW, *pstB, *scW, *scB;
    bool sc;
  } cic[8];
  int cur = 1;
  for (int s = 0; s < 8; ++s) {
    cic[s].mlpW = (const float*)d_in[cur++]; cic[s].mlpB = (const float*)d_in[cur++];
    cic[s].p2fW = (const float*)d_in[cur++]; cic[s].p2fB = (const float*)d_in[cur++];
    cic[s].preW = (const float*)d_in[cur++]; cic[s].preB = (const float*)d_in[cur++];
    cic[s].pstW = (const float*)d_in[cur++]; cic[s].pstB = (const float*)d_in[cur++];
    cic[s].sc = (DIN[s] != DOUT[s]);
    if (cic[s].sc) {
      cic[s].scW = (const float*)d_in[cur++]; cic[s].scB = (const float*)d_in[cur++];
    } else {
      cic[s].scW = nullptr; cic[s].scB = nullptr;
    }
  }
  const float* fcW = (const float*)d_in[cur++];
  const float* fcB = (const float*)d_in[cur++];
  const float* m1W = (const float*)d_in[cur++];
  const float* m1B = (const float*)d_in[cur++];
  const float* m2W = (const float*)d_in[cur++];
  const float* m2B = (const float*)d_in[cur++];
  const float* stemW = (const float*)d_in[cur++];
  const float* stemB = (const float*)d_in[cur++];

  // Workspace layout (floats).
  float* ws = (float*)d_ws;
  size_t o = 0;
  auto alloc = [&](size_t nf) { float* p = ws + o; o += nf; return p; };
  float* ptsA = alloc((size_t)BSZ * 2048 * 3);
  float* ptsB = alloc((size_t)BSZ * 2048 * 3);
  float* fA   = alloc(2097152);
  float* fB   = alloc(2097152);
  float* resB = alloc(2097152);
  float* fpre = alloc(524288);
  float* hmax = alloc(524288);
  float* h1   = alloc(10485760);
  float* hX   = alloc(1048576);
  float* gbuf = alloc((size_t)BSZ * 2048);
  float* g2   = alloc((size_t)BSZ * 512);
  int* idxFps = (int*)alloc((size_t)BSZ * 1024);
  int* idxNbr = (int*)alloc((size_t)BSZ * 1024 * NBR);
  if (o * sizeof(float) > ws_size) return;

  auto gemm = [&](const float* A, const float* W, const float* bias,
                  const float* R, float* C, int M, int Kd, int N, int act,
                  int addRes) {
    dim3 grid((M + 63) / 64, (N + 15) / 16);
    gemm_f32_wmma_k<0><<<grid, dim3(32), 0, stream>>>(A, W, bias, R, C, M, Kd,
                                                      N, act, addRes);
  };

  // Stem: f = leaky(p @ Wstem + b), (B*2048, 3) -> 32
  gemm(p_in, stemW, stemB, nullptr, fA, BSZ * 2048, 3, 32, 1, 0);

  int n_cur = 2048;
  const float* curP = p_in;
  float* nextP = ptsA;
  float* curF = fA;
  float* othF = fB;

  for (int s = 0; s < 8; ++s) {
    const int din = DIN[s], dout = DOUT[s], dh = DH[s], ng = NG[s];
    if (n_cur != ng) {
      // transition_down: FPS -> centers -> ball query -> max-pool features.
      fps_kernel<<<dim3(BSZ), dim3(FPS_T), 0, stream>>>(curP, n_cur, ng, idxFps);
      const long tg = (long)BSZ * ng;
      gather_pts_kernel<<<ceildiv_l(tg, 256), 256, 0, stream>>>(curP, idxFps,
                                                                n_cur, ng, nextP);
      const float r2 = (float)(RG[s] * RG[s]);
      ball_query_kernel<<<ceildiv_l(tg, 256), 256, 0, stream>>>(curP, nextP,
                                                                n_cur, ng, r2,
                                                                idxNbr);
      const long tp = (long)BSZ * ng * din;
      ball_pool_max_kernel<<<ceildiv_l(tp, 256), 256, 0, stream>>>(
          curF, idxNbr, n_cur, ng, din, othF);
      curP = nextP;
      nextP = (nextP == ptsA) ? ptsB : ptsA;
      float* tmp = curF; curF = othF; othF = tmp;
      n_cur = ng;
    }
    // Residual path.
    const float* resPtr;
    if (cic[s].sc) {
      gemm(curF, cic[s].scW, cic[s].scB, nullptr, resB, BSZ * ng, din, dout, 0, 0);
      resPtr = resB;
    } else {
      resPtr = curF;
    }
    // pre: leaky(f @ preW + b)
    gemm(curF, cic[s].preW, cic[s].preB, nullptr, fpre, BSZ * ng, din, dh, 1, 0);
    // KNN over current points.
    knn_kernel<<<ceildiv_l((long)BSZ * ng, 256), 256, 0, stream>>>(curP, ng, idxNbr);
    // h1 = leaky(f_ne + p2f(pfeat))
    const long th = (long)BSZ * ng * NBR * dh;
    build_h1_kernel<<<ceildiv_l(th, 256), 256, 0, stream>>>(
        fpre, curP, idxNbr, cic[s].p2fW, cic[s].p2fB, ng, dh, h1);
    // mlp GEMM + fused leaky + max-over-neighbors into hmax.
    const long tm = (long)BSZ * ng * dh;
    fill_kernel<<<ceildiv_l(tm, 256), 256, 0, stream>>>(hmax, -FLT_MAX, tm);
    {
      const int M = BSZ * ng * NBR;
      dim3 grid((M + 63) / 64, (dh + 15) / 16);
      gemm_f32_wmma_k<1><<<grid, dim3(32), 0, stream>>>(
          h1, cic[s].mlpW, cic[s].mlpB, nullptr, hmax, M, dh, dh, 1, 0);
    }
    // pst: leaky(hmax @ pstW + b + res)
    gemm(hmax, cic[s].pstW, cic[s].pstB, resPtr, othF, BSZ * ng, dh, dout, 1, 1);
    float* tmp = curF; curF = othF; othF = tmp;
  }

  // Head: mlp1 (relu) -> max/mean pool -> mlp2 (relu) -> fc.
  gemm(curF, m1W, m1B, nullptr, hX, BSZ * 64, 512, 1024, 2, 0);
  head_pool_kernel<<<ceildiv_l((long)BSZ * 1024, 256), 256, 0, stream>>>(hX, gbuf);
  gemm(gbuf, m2W, m2B, nullptr, g2, BSZ, 2048, 512, 2, 0);
  gemm(g2, fcW, fcB, nullptr, (float*)d_out, BSZ, 512, 40, 0, 0);
}